// CrossEntropyWithZLoss_69647189672133
// MI455X (gfx1250) — compile-verified
//
#include <hip/hip_runtime.h>
#include <cstdint>
#include <cstddef>

// ---------------------------------------------------------------------------
// Fused CrossEntropy + z-loss over logits[8192, 32000] fp32.
// Memory-bound (1.05 GB single pass, ~45us @ 23.3 TB/s). Single-pass online
// logsumexp; CDNA5 async-to-LDS 4-deep ring buffering (ASYNCcnt) to maximize
// per-wave memory-level parallelism on the HBM stream.
// ---------------------------------------------------------------------------

#define IGNORE_INDEX (-100)

constexpr int   N_ROWS  = 8192;
constexpr int   VCLS    = 32000;
constexpr int   BLOCK   = 320;           // 10 wave32s
constexpr int   NWAVES  = BLOCK / 32;    // 10
constexpr int   VEC4    = VCLS / 4;      // 8000 float4 per row
constexpr int   CHUNKS  = VEC4 / BLOCK;  // 25 chunks, exact
constexpr int   PIPE    = 4;             // async ring depth (power of 2)
constexpr float LSE2_SCALE = 1e-4f;
constexpr float SMOOTH_EPS = 0.1f;

__global__ void cez_zero_ws(double* ws) {
    ws[0] = 0.0;   // sum of (ce + z) over valid rows
    ws[1] = 0.0;   // n_valid
}

__global__ __launch_bounds__(BLOCK)
void cez_rows_kernel(const float* __restrict__ logits,
                     const int*   __restrict__ targets,
                     double*      __restrict__ ws)
{
    __shared__ float4 sbuf[PIPE * BLOCK];       // ring: 4 x 5120 B = 20 KB
    __shared__ float  red_m[NWAVES];
    __shared__ float  red_s[NWAVES];
    __shared__ float  red_x[NWAVES];

    const int tid = threadIdx.x;
    const int row = blockIdx.x;
    const float* rowp = logits + (size_t)row * VCLS;

    // Wave-relative LDS byte offset of this lane's slot in ring buffer 0
    // (low 32 bits of the flat shared-memory address are the LDS offset).
    const unsigned lane_lds = (unsigned)(uintptr_t)&sbuf[tid];

    // Online logsumexp state (+ plain sum for label-smoothing mean).
    float m  = -3.402823466e38f;   // running max
    float s  = 0.0f;               // running sum of exp(x - m)
    float sx = 0.0f;               // running sum of x

    auto issue = [&](int c) {
        unsigned dst  = lane_lds + (unsigned)((c & (PIPE - 1)) * BLOCK * 16);
        unsigned goff = (unsigned)((c * BLOCK + tid) * 16);
        asm volatile("global_load_async_to_lds_b128 %0, %1, %2"
                     :: "v"(dst), "v"(goff), "s"(rowp)
                     : "memory");
    };

    auto process = [&](int c) {
        // Each lane reads back exactly the 16B it async-loaded (no barrier
        // needed: producer and consumer are the same lane's LDS slot).
        float4 x = sbuf[(c & (PIPE - 1)) * BLOCK + tid];
        float cm = fmaxf(fmaxf(x.x, x.y), fmaxf(x.z, x.w));
        if (cm > m) { s *= __expf(m - cm); m = cm; }
        s += __expf(x.x - m) + __expf(x.y - m) +
             __expf(x.z - m) + __expf(x.w - m);
        sx += (x.x + x.y) + (x.z + x.w);
    };

    // ---- prime the 4-deep pipeline --------------------------------------
    issue(0);
    issue(1);
    issue(2);

    // ---- steady state: 4 chunk loads in flight per wave ------------------
    for (int c = 0; c < CHUNKS - 3; ++c) {
        issue(c + 3);
        asm volatile("s_wait_asynccnt 0x3" ::: "memory");  // oldest done
        process(c);
    }
    // ---- drain tail (compile-time wait immediates) -----------------------
    asm volatile("s_wait_asynccnt 0x2" ::: "memory");
    process(CHUNKS - 3);
    asm volatile("s_wait_asynccnt 0x1" ::: "memory");
    process(CHUNKS - 2);
    asm volatile("s_wait_asynccnt 0x0" ::: "memory");
    process(CHUNKS - 1);

    // ---- wave32 reduction of (m, s, sx) ---------------------------------
    #pragma unroll
    for (int off = 16; off > 0; off >>= 1) {
        float om = __shfl_xor(m,  off, 32);
        float os = __shfl_xor(s,  off, 32);
        float ox = __shfl_xor(sx, off, 32);
        float nm = fmaxf(m, om);
        s  = s * __expf(m - nm) + os * __expf(om - nm);
        m  = nm;
        sx += ox;
    }

    const int wave = tid >> 5;
    if ((tid & 31) == 0) {
        red_m[wave] = m;
        red_s[wave] = s;
        red_x[wave] = sx;
    }
    __syncthreads();

    // ---- final per-row math on thread 0 ---------------------------------
    if (tid == 0) {
        float M = red_m[0], S = red_s[0], X = red_x[0];
        #pragma unroll
        for (int w = 1; w < NWAVES; ++w) {
            float om = red_m[w], os = red_s[w];
            float nm = fmaxf(M, om);
            S = S * __expf(M - nm) + os * __expf(om - nm);
            M = nm;
            X += red_x[w];
        }
        float lse = M + __logf(S);

        int t = targets[row];
        if (t != IGNORE_INDEX) {
            float tl     = rowp[t];                         // one extra load
            float nll    = lse - tl;
            float smooth = lse - X * (1.0f / (float)VCLS);
            float ce     = (1.0f - SMOOTH_EPS) * nll + SMOOTH_EPS * smooth;
            float tot    = ce + LSE2_SCALE * lse * lse;
            atomicAdd(&ws[0], (double)tot);
            atomicAdd(&ws[1], 1.0);
        }
    }
}

__global__ void cez_finalize(const double* __restrict__ ws,
                             float* __restrict__ out)
{
    double nv = ws[1];
    out[0] = (float)(nv > 0.0 ? ws[0] / nv : 0.0);
}

extern "C" void kernel_launch(void* const* d_in, const int* in_sizes, int n_in,
                              void* d_out, int out_size, void* d_ws, size_t ws_size,
                              hipStream_t stream)
{
    (void)in_sizes; (void)n_in; (void)out_size; (void)ws_size;

    const float* logits  = (const float*)d_in[0];
    const int*   targets = (const int*)d_in[1];
    double*      ws      = (double*)d_ws;     // [0]=loss sum, [1]=n_valid
    float*       out     = (float*)d_out;

    cez_zero_ws<<<1, 1, 0, stream>>>(ws);
    cez_rows_kernel<<<N_ROWS, BLOCK, 0, stream>>>(logits, targets, ws);
    cez_finalize<<<1, 1, 0, stream>>>(ws, out);
}